// R1Pool_19902878449851
// MI455X (gfx1250) — compile-verified
//
#include <hip/hip_runtime.h>
#include <stdint.h>

// Problem shape (fixed by the reference): B=8, C=64, H=256, W=256
#define PH       256
#define PW       256
#define NPLANES  512          // B*C
#define PFD      16           // async prefetch depth (rows), power of two

typedef float v4f __attribute__((ext_vector_type(4)));   // native vec4 (b128)

// ---------------------------------------------------------------------------
// u64 key: (monotone float bits)<<16 | col<<8 | row.
// Total order (value, col, row), "larger wins" == the reference's tie rules,
// so the whole 2D quadrant-argmax is a commutative/associative u64 max.
// ---------------------------------------------------------------------------
__device__ __forceinline__ unsigned long long make_key(float g, int row, int col) {
    unsigned u = __float_as_uint(g);
    u = (u & 0x80000000u) ? ~u : (u | 0x80000000u);       // monotone float->u32
    return ((unsigned long long)u << 16) | ((unsigned)col << 8) | (unsigned)row;
}

__device__ __forceinline__ unsigned long long kmax(unsigned long long a,
                                                   unsigned long long b) {
    return a > b ? a : b;
}

// DPP row_shr step on a u64 (two 32-bit DPP moves, identity 0 shifted in).
template <int CTRL>
__device__ __forceinline__ unsigned long long dppshr_u64(unsigned long long v) {
    int lo = __builtin_amdgcn_update_dpp(0, (int)(unsigned)v,         CTRL, 0xf, 0xf, true);
    int hi = __builtin_amdgcn_update_dpp(0, (int)(unsigned)(v >> 32), CTRL, 0xf, 0xf, true);
    return ((unsigned long long)(unsigned)hi << 32) | (unsigned)lo;
}

// ---- CDNA5 async global->LDS (inline asm; tracked by ASYNCcnt) ------------
__device__ __forceinline__ void async_b128(const float* gaddr, unsigned lds_off) {
    asm volatile("global_load_async_to_lds_b128 %0, %1, off"
                 :: "v"(lds_off), "v"(gaddr) : "memory");
}
#define WAIT_ASYNC(n)  asm volatile("s_wait_asynccnt " #n ::: "memory")
#define WAIT_DS0()     asm volatile("s_wait_dscnt 0"      ::: "memory")

__global__ __launch_bounds__(32)
void r1pool_kernel(const float* __restrict__ x,
                   const float* __restrict__ guide,
                   float* __restrict__ out) {
    const int p    = blockIdx.x;                 // one wave == one plane
    const int l    = threadIdx.x;                // lane: owns columns 8l..8l+7
    const size_t base = (size_t)p * (PH * PW);

    const float* gplane = guide + base;
    const float* xplane = x + base;
    float*       oplane = out + base;

    __shared__ __align__(16) float ring[PFD][PW];   // 16 KB async ring buffer

    // ---- prime the async pipeline: DMA rows 0..PFD-1 into the ring --------
    const int c0 = l << 3;                       // first column owned by lane
    #pragma unroll
    for (int r = 0; r < PFD; ++r) {
        const float* src = gplane + r * PW + c0;
        unsigned dst = (unsigned)(uintptr_t)&ring[r][c0];
        async_b128(src,     dst);
        async_b128(src + 4, dst + 16);
    }

    unsigned long long carr[8];                  // per-column quadrant carries
    #pragma unroll
    for (int k = 0; k < 8; ++k) carr[k] = 0ull;

    float pend[8];                               // gathered row, stored next iter

    for (int i = 0; i < PH; ++i) {
        const int slot = i & (PFD - 1);

        // Wait for this row's pair of async ops (in-order completion).
        if (i + PFD < PH) { WAIT_ASYNC(30); }    // 2*(PFD-1) still in flight
        else              { WAIT_ASYNC(0);  }    // drain in the tail

        v4f ga = *(const v4f*)&ring[slot][c0];
        v4f gb = *(const v4f*)&ring[slot][c0 + 4];

        // Build keys + local inclusive prefix over the lane's 8 columns.
        unsigned long long h[8];
        h[0] = make_key(ga.x, i, c0 + 0);
        h[1] = kmax(h[0], make_key(ga.y, i, c0 + 1));
        h[2] = kmax(h[1], make_key(ga.z, i, c0 + 2));
        h[3] = kmax(h[2], make_key(ga.w, i, c0 + 3));
        h[4] = kmax(h[3], make_key(gb.x, i, c0 + 4));
        h[5] = kmax(h[4], make_key(gb.y, i, c0 + 5));
        h[6] = kmax(h[5], make_key(gb.z, i, c0 + 6));
        h[7] = kmax(h[6], make_key(gb.w, i, c0 + 7));

        // Ring slot consumed -> safe to DMA row i+PFD into the same slot.
        if (i + PFD < PH) {
            WAIT_DS0();
            const float* src = gplane + (size_t)(i + PFD) * PW + c0;
            unsigned dst = (unsigned)(uintptr_t)&ring[slot][c0];
            async_b128(src,     dst);
            async_b128(src + 4, dst + 16);
        }

        // Wave-level inclusive max-scan of lane totals: 4 DPP row_shr steps
        // (within 16-lane rows), then merge lane-15 total into the upper half.
        unsigned long long s = h[7];
        s = kmax(s, dppshr_u64<0x111>(s));
        s = kmax(s, dppshr_u64<0x112>(s));
        s = kmax(s, dppshr_u64<0x114>(s));
        s = kmax(s, dppshr_u64<0x118>(s));
        unsigned lo15 = (unsigned)__builtin_amdgcn_readlane((int)(unsigned)s, 15);
        unsigned hi15 = (unsigned)__builtin_amdgcn_readlane((int)(unsigned)(s >> 32), 15);
        unsigned long long t15 = ((unsigned long long)hi15 << 32) | lo15;
        if (l >= 16) s = kmax(s, t15);           // full-wave inclusive scan

        // Exclusive-of-this-lane prefix: shift by one lane; patch lane 16.
        unsigned long long e = dppshr_u64<0x111>(s);
        if (l == 16) e = t15;

        // Finalize each column: S = max(carry, wave-left prefix, local prefix);
        // decode dominant flat index and gather from x (L2-resident plane).
        float cur[8];
        #pragma unroll
        for (int k = 0; k < 8; ++k) {
            unsigned long long S = kmax(carr[k], kmax(e, h[k]));
            carr[k] = S;
            int dom = (int)(((S & 0xFFull) << 8) | ((S >> 8) & 0xFFull));
            cur[k] = xplane[dom];
        }

        // Store the PREVIOUS row's gathered values (keeps L2 gather latency
        // off the scan's critical path), as two non-temporal b128 stores.
        if (i > 0) {
            v4f oa = { pend[0], pend[1], pend[2], pend[3] };
            v4f ob = { pend[4], pend[5], pend[6], pend[7] };
            __builtin_nontemporal_store(oa, (v4f*)&oplane[((i - 1) << 8) + c0]);
            __builtin_nontemporal_store(ob, (v4f*)&oplane[((i - 1) << 8) + c0 + 4]);
        }
        #pragma unroll
        for (int k = 0; k < 8; ++k) pend[k] = cur[k];
    }

    // Flush the last row.
    v4f oa = { pend[0], pend[1], pend[2], pend[3] };
    v4f ob = { pend[4], pend[5], pend[6], pend[7] };
    __builtin_nontemporal_store(oa, (v4f*)&oplane[((PH - 1) << 8) + c0]);
    __builtin_nontemporal_store(ob, (v4f*)&oplane[((PH - 1) << 8) + c0 + 4]);
}

extern "C" void kernel_launch(void* const* d_in, const int* in_sizes, int n_in,
                              void* d_out, int out_size, void* d_ws, size_t ws_size,
                              hipStream_t stream) {
    const float* x     = (const float*)d_in[0];
    const float* guide = (const float*)d_in[1];
    float*       out   = (float*)d_out;

    dim3 grid(NPLANES);   // 512 independent planes, one wave32 each
    dim3 block(32);       // barrier-free: wave-private async + DPP scan
    hipLaunchKernelGGL(r1pool_kernel, grid, block, 0, stream, x, guide, out);
}